// GramVolumeCalculator_62637803045021
// MI455X (gfx1250) — compile-verified
//
#include <hip/hip_runtime.h>
#include <math.h>

// ---------------------------------------------------------------------------
// GramVolumeCalculator for MI455X (gfx1250, wave32).
// Memory-bound streaming kernel (768 MB @ 23.3 TB/s ~= 33 us floor).
// Gram dot products computed on the matrix pipe via V_WMMA_F32_16X16X4_F32:
// one wave owns 16 batch rows; per K-step of 4 it loads one float2 per input
// matrix per lane (layout serves as BOTH the A and B operand for this shape)
// and issues 6 WMMAs (n1,n2,n3,d12,d13,d23). Diagonals of the 16x16
// accumulators are the per-row dot products.
// ---------------------------------------------------------------------------

typedef float v2f __attribute__((ext_vector_type(2)));
typedef float v8f __attribute__((ext_vector_type(8)));

#define DIMK 1024
#define ROWS_PER_WAVE 16

#if defined(__has_builtin)
#if __has_builtin(__builtin_amdgcn_wmma_f32_16x16x4_f32)
#define HAVE_WMMA_F32X4 1
#endif
#endif

__device__ __forceinline__ v8f wmma_acc(v2f a, v2f b, v8f c) {
#ifdef HAVE_WMMA_F32X4
  // 8 args: (neg_a, A, neg_b, B, c_mod, C, reuse_a, reuse_b)
  return __builtin_amdgcn_wmma_f32_16x16x4_f32(
      false, a, false, b, (short)0, c, false, false);
#else
  // Compile-keeping fallback (histogram will show wmma=0 if builtin missing).
  c[0] += a[0] * b[0] + a[1] * b[1];
  return c;
#endif
}

// Select acc[lane & 7] with unrolled compile-time indices (lowers to cndmasks,
// no scratch). Per the 16x16 f32 C/D layout, after this select the diagonal
// element for row i sits on lane i (i<8) or lane i+16 (i>=8).
__device__ __forceinline__ float pick_diag(v8f acc, int lane) {
  float d = acc[0];
  const int c = lane & 7;
#pragma unroll
  for (int j = 1; j < 8; ++j) {
    if (c == j) d = acc[j];
  }
  return d;
}

__global__ __launch_bounds__(256) void gram_volume_wmma_kernel(
    const float* __restrict__ f1, const float* __restrict__ f2,
    const float* __restrict__ f3, float* __restrict__ out, int B) {
  const int lane = threadIdx.x & 31;
  const int wave = threadIdx.x >> 5;
  const int tile = blockIdx.x * (blockDim.x >> 5) + wave;
  const int rowbase = tile * ROWS_PER_WAVE;
  if (rowbase >= B) return;  // wave-uniform: EXEC stays all-1s for WMMA

  // Operand layout for V_WMMA_F32_16X16X4_F32 (A 16x4 / B 4x16, f32):
  //   lane l -> row (l & 15), K elements {2*(l>>4), 2*(l>>4)+1} of the step.
  const int r = lane & 15;
  const int h = lane >> 4;
  const size_t base = (size_t)(rowbase + r) * DIMK + (size_t)(h << 1);

  const float* p1 = f1 + base;
  const float* p2 = f2 + base;
  const float* p3 = f3 + base;

  v8f a11 = {}, a22 = {}, a33 = {}, a12 = {}, a13 = {}, a23 = {};

#pragma unroll 4
  for (int kk = 0; kk < DIMK; kk += 4) {
    v2f x1 = *(const v2f*)p1;
    v2f x2 = *(const v2f*)p2;
    v2f x3 = *(const v2f*)p3;
    p1 += 4;
    p2 += 4;
    p3 += 4;
    // Same registers serve as A (16x4) and B (4x16): Gram = A * A^T.
    a11 = wmma_acc(x1, x1, a11);
    a22 = wmma_acc(x2, x2, a22);
    a33 = wmma_acc(x3, x3, a33);
    a12 = wmma_acc(x1, x2, a12);
    a13 = wmma_acc(x1, x3, a13);
    a23 = wmma_acc(x2, x3, a23);
  }

  const float n1 = pick_diag(a11, lane);
  const float n2 = pick_diag(a22, lane);
  const float n3 = pick_diag(a33, lane);
  const float d12 = pick_diag(a12, lane);
  const float d13 = pick_diag(a13, lane);
  const float d23 = pick_diag(a23, lane);

  // Owner lanes: {0..7} hold rows 0..7, {24..31} hold rows 8..15.
  const bool owner = (lane < 8) | (lane >= 24);
  if (owner) {
    const int i = (lane < 8) ? lane : (lane - 16);

    // Fold L2 normalization analytically: F.normalize(x) = x / max(||x||,1e-12)
    const float inv1 = 1.0f / fmaxf(sqrtf(n1), 1e-12f);
    const float inv2 = 1.0f / fmaxf(sqrtf(n2), 1e-12f);
    const float inv3 = 1.0f / fmaxf(sqrtf(n3), 1e-12f);

    const float g11 = n1 * inv1 * inv1;
    const float g22 = n2 * inv2 * inv2;
    const float g33 = n3 * inv3 * inv3;
    const float g12 = d12 * inv1 * inv2;
    const float g13 = d13 * inv1 * inv3;
    const float g23 = d23 * inv2 * inv3;

    float det = g11 * g22 * g33 + 2.0f * g12 * g13 * g23 -
                g11 * g23 * g23 - g22 * g13 * g13 - g33 * g12 * g12;
    det = fmaxf(det, 1e-8f);
    const float vol = sqrtf(det);
    // sigmoid(-10*vol + 5) = 1 / (1 + exp(10*vol - 5))
    const float score = 1.0f / (1.0f + __expf(vol * 10.0f - 5.0f));

    const int row = rowbase + i;
    if (row < B) {
      out[row] = vol;        // volume
      out[B + row] = score;  // alignment_score
    }
  }
}

extern "C" void kernel_launch(void* const* d_in, const int* in_sizes, int n_in,
                              void* d_out, int out_size, void* d_ws,
                              size_t ws_size, hipStream_t stream) {
  (void)n_in;
  (void)d_ws;
  (void)ws_size;
  (void)out_size;
  const float* f1 = (const float*)d_in[0];
  const float* f2 = (const float*)d_in[1];
  const float* f3 = (const float*)d_in[2];
  float* out = (float*)d_out;

  const int B = in_sizes[0] / DIMK;                 // 65536
  const int tiles = (B + ROWS_PER_WAVE - 1) / ROWS_PER_WAVE;  // 4096
  const int wavesPerBlock = 256 / 32;               // 8
  const int blocks = (tiles + wavesPerBlock - 1) / wavesPerBlock;  // 512

  gram_volume_wmma_kernel<<<blocks, 256, 0, stream>>>(f1, f2, f3, out, B);
}